// ConvolutionSparseLayer_17437567222235
// MI455X (gfx1250) — compile-verified
//
#include <hip/hip_runtime.h>
#include <stdint.h>

// ---------------------------------------------------------------------------
// GCN sparse layer: out = relu( A_sp @ ( X_sp @ W ) )   (MI455X / gfx1250)
//   * CSR build: count -> WMMA triangular-matmul scan -> scatter permute.
//     (scan-by-matrix: Z = X@U + L@T on v_wmma_f32_16x16x4_f32; exact since
//      all values are integer-valued f32 < 2^24)
//   * SpMM = one-wave32-per-row gather, register accumulators, zero output
//     atomics (vs 1.64e9 f32 atomics in the naive scatter form).
//   * xw (102.4 MB) stays resident in the 192 MB L2 -> SpMM2 gathers never
//     touch HBM; final output uses non-temporal stores to avoid evicting it.
//   * SpMM1 stages 256KB W column-slabs into LDS with the Tensor Data Mover
//     (tensor_load_to_lds + s_wait_tensorcnt); 1024-thread blocks = 32 waves
//     owning one WGP and its LDS.
// ---------------------------------------------------------------------------

#define N_NODES   100000
#define IN_DIM    1024
#define OUT_DIM   256
#define SLABS     4
#define SLAB_COLS (OUT_DIM / SLABS)                          // 64 cols/slab
#define LDS_BYTES (SLAB_COLS * IN_DIM * (int)sizeof(float))  // 262144 bytes
#define SCAN_TILE 256
#define N_CHUNKS  ((N_NODES + SCAN_TILE - 1) / SCAN_TILE)    // 391

typedef __attribute__((ext_vector_type(4))) unsigned int u32x4;
typedef __attribute__((ext_vector_type(8))) int          i32x8;
typedef __attribute__((ext_vector_type(4))) int          i32x4;
typedef __attribute__((ext_vector_type(2))) float        f32x2;
typedef __attribute__((ext_vector_type(4))) float        f32x4;
typedef __attribute__((ext_vector_type(8))) float        f32x8;

#if defined(__has_builtin)
#if __has_builtin(__builtin_amdgcn_tensor_load_to_lds)
#define HAVE_TDM 1
#endif
#if __has_builtin(__builtin_amdgcn_wmma_f32_16x16x4_f32)
#define HAVE_WMMA 1
#endif
#endif
#ifndef HAVE_TDM
#define HAVE_TDM 0
#endif
#ifndef HAVE_WMMA
#define HAVE_WMMA 0
#endif

// --------------------------- utility kernels -------------------------------

__global__ void zero_u32(unsigned* __restrict__ p, int n) {
  int i = blockIdx.x * blockDim.x + threadIdx.x;
  if (i < n) p[i] = 0u;
}

__global__ void count_rows(const int* __restrict__ rows,
                           unsigned* __restrict__ cnt, int nnz) {
  int i = blockIdx.x * blockDim.x + threadIdx.x;
  if (i < nnz) atomicAdd(&cnt[rows[i]], 1u);
}

// ---- Phase P: per-256-tile exclusive scan, one wave32 per tile ------------
// WMMA path: view the tile as X (16x16, element i = 16*m + n).
//   Y = X @ U        (U[c'][c] = c'<=c)  -> within-row inclusive scans
//   Z = L @ T + Y    (L[r][r'] = r'<r, T[r'][*] = Y[r'][15]) -> full scan
// 8 chained v_wmma_f32_16x16x4_f32 per tile; integer-exact in f32.
__global__ void __launch_bounds__(32)
scan_partial_k(const unsigned* __restrict__ cnt_x,
               unsigned* __restrict__ rp_x,
               const unsigned* __restrict__ cnt_a,
               unsigned* __restrict__ rp_a,
               unsigned* __restrict__ chunkSums,
               int n, int nChunks) {
  const int mm = (int)blockIdx.x / nChunks;
  const int c  = (int)blockIdx.x % nChunks;
  const unsigned* cnt = mm ? cnt_a : cnt_x;
  unsigned* rp = mm ? rp_a : rp_x;
  const int base = c * SCAN_TILE;
  const int lane = threadIdx.x;       // 0..31, full wave

#if HAVE_WMMA
  const int nn    = lane & 15;        // N index (and A-operand M index)
  const int khalf = lane >> 4;        // selects K 0/1 vs 2/3 within a chunk
  __shared__ float tot[16];

  f32x8 acc = {0.f, 0.f, 0.f, 0.f, 0.f, 0.f, 0.f, 0.f};
  // Phase 1: Y = X @ U  (A = X columns, B = U rows, both per K=4 chunk)
  for (int k4 = 0; k4 < 4; ++k4) {
    f32x2 A, B;
    for (int v = 0; v < 2; ++v) {
      const int kk  = 4 * k4 + v + 2 * khalf;   // K index in [0,16)
      const int i   = base + 16 * nn + kk;      // A[m=nn][k=kk] = X[nn][kk]
      A[v] = (i < n) ? (float)cnt[i] : 0.f;
      B[v] = (kk <= nn) ? 1.f : 0.f;            // U[kk][n=nn]
    }
    acc = __builtin_amdgcn_wmma_f32_16x16x4_f32(false, A, false, B,
                                                (short)0, acc, false, false);
  }
  // Row totals Y[m][15] live in lanes 15 (m=0..7) and 31 (m=8..15).
  if (nn == 15) {
    for (int g = 0; g < 8; ++g) tot[g + 8 * khalf] = acc[g];
  }
  __syncthreads();  // single-wave workgroup: barrier degenerates to S_NOP
  // Phase 2: Z = L_strict @ T + Y
  for (int k4 = 0; k4 < 4; ++k4) {
    f32x2 A, B;
    for (int v = 0; v < 2; ++v) {
      const int kk = 4 * k4 + v + 2 * khalf;    // r' index
      A[v] = (kk < nn) ? 1.f : 0.f;             // L[m=nn][r'=kk]
      B[v] = tot[kk];                           // T[r'=kk][n] (uniform in n)
    }
    acc = __builtin_amdgcn_wmma_f32_16x16x4_f32(false, A, false, B,
                                                (short)0, acc, false, false);
  }
  // D layout: VGPR g -> M = g + 8*khalf, N = nn. Write exclusive values.
  for (int g = 0; g < 8; ++g) {
    const int i = base + 16 * (g + 8 * khalf) + nn;
    if (i < n) {
      const unsigned incl = (unsigned)(acc[g] + 0.5f);
      rp[i] = incl - cnt[i];
    }
  }
  if (lane == 31) {  // Z[15][15] = tile total
    chunkSums[mm * nChunks + c] = (unsigned)(acc[7] + 0.5f);
  }
#else
  // Fallback (and host-pass body): lane-serial + shfl wave scan.
  unsigned ex[8];
  unsigned run = 0;
  const int lbase = base + lane * 8;
  for (int t = 0; t < 8; ++t) {
    const int i = lbase + t;
    const unsigned v = (i < n) ? cnt[i] : 0u;
    ex[t] = run;
    run += v;
  }
  unsigned inc = run;
  for (int off = 1; off < 32; off <<= 1) {
    const unsigned u = __shfl_up(inc, off, 32);
    if (lane >= off) inc += u;
  }
  const unsigned waveExcl = inc - run;
  for (int t = 0; t < 8; ++t) {
    const int i = lbase + t;
    if (i < n) rp[i] = waveExcl + ex[t];
  }
  if (lane == 31) chunkSums[mm * nChunks + c] = inc;
#endif
}

// Phase M: one block scans the 391 chunk sums for each matrix,
// writes chunk offsets and the final rowptr[N] totals.
__global__ void scan_sums(const unsigned* __restrict__ chunkSums,
                          unsigned* __restrict__ chunkOffs,
                          unsigned* __restrict__ rp_x,
                          unsigned* __restrict__ rp_a,
                          int n, int nChunks) {
  __shared__ unsigned sd[512];
  const int tid = threadIdx.x;
  for (int m = 0; m < 2; ++m) {
    const unsigned v = (tid < nChunks) ? chunkSums[m * nChunks + tid] : 0u;
    sd[tid] = v;
    __syncthreads();
    for (int off = 1; off < 512; off <<= 1) {
      unsigned t = (tid >= off) ? sd[tid - off] : 0u;
      __syncthreads();
      sd[tid] += t;
      __syncthreads();
    }
    if (tid < nChunks) chunkOffs[m * nChunks + tid] = sd[tid] - v;
    if (tid == 511) (m ? rp_a : rp_x)[n] = sd[511];
    __syncthreads();
  }
}

// Phase F: add chunk offsets; also initialize the scatter cursors.
__global__ void scan_finalize(unsigned* __restrict__ rp_x,
                              unsigned* __restrict__ cur_x,
                              unsigned* __restrict__ rp_a,
                              unsigned* __restrict__ cur_a,
                              const unsigned* __restrict__ chunkOffs,
                              int n, int nChunks) {
  const int m = (int)blockIdx.x / nChunks;
  const int c = (int)blockIdx.x % nChunks;
  unsigned* rp  = m ? rp_a : rp_x;
  unsigned* cur = m ? cur_a : cur_x;
  const int i = c * SCAN_TILE + (int)threadIdx.x;
  if (i < n) {
    const unsigned val = rp[i] + chunkOffs[m * nChunks + c];
    rp[i]  = val;
    cur[i] = val;
  }
}

// Permute COO (col,val) pairs into CSR order (packed 64-bit entries:
// low 32 = column index, high 32 = f32 value bits).
__global__ void scatter_csr(const int* __restrict__ rows,
                            const int* __restrict__ cols,
                            const float* __restrict__ vals,
                            unsigned* __restrict__ cursor,
                            long long* __restrict__ entries, int nnz) {
  int i = blockIdx.x * blockDim.x + threadIdx.x;
  if (i < nnz) {
    const int r = rows[i];
    const unsigned pos = atomicAdd(&cursor[r], 1u);
    const unsigned long long packed =
        (unsigned long long)(unsigned)cols[i] |
        ((unsigned long long)(unsigned)__float_as_int(vals[i]) << 32);
    entries[pos] = (long long)packed;
  }
}

// ------------------------------ SpMM 1 -------------------------------------
// xw[row, slab*64 : +64] = sum_j val_j * W[col_j, slab cols]
// One block = one WGP: 32 wave32s + 256KB LDS W-slab staged by the TDM.

__global__ void __launch_bounds__(1024, 1)
spmm1_xw(const unsigned* __restrict__ rowptr,
         const long long* __restrict__ entries,
         const float* __restrict__ W,
         float* __restrict__ xw, int nrows) {
  extern __shared__ float ldsW[];  // SLAB_COLS * IN_DIM floats = 256KB
  const int slab = (int)(blockIdx.x) % SLABS;
  const int bos  = (int)(blockIdx.x) / SLABS;

#if HAVE_TDM
  if (threadIdx.x < 32) {  // wave 0 issues the DMA; TDM ignores EXEC
    const uint64_t gaddr = (uint64_t)(uintptr_t)(W + (size_t)slab * SLAB_COLS);
    const unsigned lds_addr = (unsigned)(uintptr_t)(&ldsW[0]);
    // D# group 0: count=1 valid, lds_addr, 57-bit global addr, type=2.
    u32x4 g0;
    g0[0] = 1u;
    g0[1] = lds_addr;
    g0[2] = (unsigned)(gaddr & 0xFFFFFFFFull);
    g0[3] = (unsigned)((gaddr >> 32) & 0x1FFFFFFull) | (2u << 30);
    // D# group 1: data_size=4B, tensor 64 x 1024 (row stride 256 elements),
    // tile 64 x 1024 -> one 256KB 2D DMA into LDS.
    i32x8 g1 = {0, 0, 0, 0, 0, 0, 0, 0};
    g1[0] = (int)(2u << 16);                        // data_size = 4 bytes
    g1[1] = (int)((SLAB_COLS & 0xFFFF) << 16);      // tensor_dim0 lo16
    g1[2] = (int)((IN_DIM & 0xFFFF) << 16);         // tdim0 hi=0 | tdim1 lo16
    g1[3] = (int)((SLAB_COLS & 0xFFFF) << 16);      // tdim1 hi=0 | tile_dim0
    g1[4] = (int)(IN_DIM & 0xFFFF);                 // tile_dim1 | tile_dim2=0
    g1[5] = (int)OUT_DIM;                           // tensor_dim0_stride
    i32x4 gz4 = {0, 0, 0, 0};
#if defined(__clang_major__) && (__clang_major__ >= 23)
    i32x8 gz8 = {0, 0, 0, 0, 0, 0, 0, 0};
    __builtin_amdgcn_tensor_load_to_lds(g0, g1, gz4, gz4, gz8, 0);
#else
    __builtin_amdgcn_tensor_load_to_lds(g0, g1, gz4, gz4, 0);
#endif
    __builtin_amdgcn_s_wait_tensorcnt((short)0);
  }
#else
  for (int idx = threadIdx.x; idx < SLAB_COLS * IN_DIM; idx += blockDim.x) {
    const int k = idx >> 6, c = idx & (SLAB_COLS - 1);
    ldsW[idx] = W[(size_t)k * OUT_DIM + slab * SLAB_COLS + c];
  }
#endif
  __syncthreads();

  const int lane = threadIdx.x & 31;
  const int waveInBlock = threadIdx.x >> 5;
  const int wavesPerSlab = ((int)gridDim.x / SLABS) * 32;
  const float2* __restrict__ lds2 = (const float2*)ldsW;

  for (int row = bos * 32 + waveInBlock; row < nrows; row += wavesPerSlab) {
    const unsigned s = rowptr[row], e = rowptr[row + 1];
    float2 acc; acc.x = 0.f; acc.y = 0.f;
    for (unsigned j = s; j < e; ++j) {
      __builtin_prefetch(entries + j + 16, 0, 0);   // global_prefetch_b8
      const unsigned long long en =
          (unsigned long long)entries[j];           // wave-uniform broadcast
      const int   col = (int)(unsigned)(en & 0xFFFFFFFFull);
      const float v   = __int_as_float((int)(unsigned)(en >> 32));
      const float2 w = lds2[col * (SLAB_COLS / 2) + lane];  // conflict-free
      acc.x = fmaf(v, w.x, acc.x);
      acc.y = fmaf(v, w.y, acc.y);
    }
    *(float2*)(xw + (size_t)row * OUT_DIM + slab * SLAB_COLS + (lane << 1)) = acc;
  }
}

// ------------------------------ SpMM 2 + ReLU ------------------------------
// One wave32 per output row; each lane keeps 8 f32 accumulators (256 cols).
// xw gathers are served from the 192MB L2; entries stream is non-temporal;
// final stores are non-temporal (output is never re-read).

__device__ __forceinline__ void fma8(f32x4& a0, f32x4& a1, float v,
                                     const float* __restrict__ base) {
  const f32x4 a = *(const f32x4*)base;
  const f32x4 b = *(const f32x4*)(base + 4);
  a0.x = fmaf(v, a.x, a0.x); a0.y = fmaf(v, a.y, a0.y);
  a0.z = fmaf(v, a.z, a0.z); a0.w = fmaf(v, a.w, a0.w);
  a1.x = fmaf(v, b.x, a1.x); a1.y = fmaf(v, b.y, a1.y);
  a1.z = fmaf(v, b.z, a1.z); a1.w = fmaf(v, b.w, a1.w);
}

__global__ void __launch_bounds__(256)
spmm2_relu(const unsigned* __restrict__ rowptr,
           const long long* __restrict__ entries,
           const float* __restrict__ xw,
           float* __restrict__ out, int nrows) {
  const int lane = threadIdx.x & 31;
  const int row  = (int)blockIdx.x * (blockDim.x >> 5) + (threadIdx.x >> 5);
  if (row >= nrows) return;
  const unsigned s = rowptr[row], e = rowptr[row + 1];
  f32x4 acc0 = {0.f, 0.f, 0.f, 0.f};
  f32x4 acc1 = {0.f, 0.f, 0.f, 0.f};
  unsigned j = s;
  // 2x unroll: two independent 1KB row-gathers in flight per iteration.
  for (; j + 1 < e; j += 2) {
    __builtin_prefetch(entries + j + 16, 0, 0);
    const unsigned long long e0 =
        (unsigned long long)__builtin_nontemporal_load(entries + j);
    const unsigned long long e1 =
        (unsigned long long)__builtin_nontemporal_load(entries + j + 1);
    const float* __restrict__ p0 =
        xw + (size_t)(unsigned)(e0 & 0xFFFFFFFFull) * OUT_DIM + (lane << 3);
    const float* __restrict__ p1 =
        xw + (size_t)(unsigned)(e1 & 0xFFFFFFFFull) * OUT_DIM + (lane << 3);
    fma8(acc0, acc1, __int_as_float((int)(unsigned)(e0 >> 32)), p0);
    fma8(acc0, acc1, __int_as_float((int)(unsigned)(e1 >> 32)), p1);
  }
  if (j < e) {
    const unsigned long long e0 =
        (unsigned long long)__builtin_nontemporal_load(entries + j);
    const float* __restrict__ p0 =
        xw + (size_t)(unsigned)(e0 & 0xFFFFFFFFull) * OUT_DIM + (lane << 3);
    fma8(acc0, acc1, __int_as_float((int)(unsigned)(e0 >> 32)), p0);
  }
  acc0.x = fmaxf(acc0.x, 0.f); acc0.y = fmaxf(acc0.y, 0.f);
  acc0.z = fmaxf(acc0.z, 0.f); acc0.w = fmaxf(acc0.w, 0.f);
  acc1.x = fmaxf(acc1.x, 0.f); acc1.y = fmaxf(acc1.y, 0.f);
  acc1.z = fmaxf(acc1.z, 0.f); acc1.w = fmaxf(acc1.w, 0.f);
  float* o = out + (size_t)row * OUT_DIM + (lane << 3);
  __builtin_nontemporal_store(acc0, (f32x4*)o);
  __builtin_nontemporal_store(acc1, (f32x4*)o + 1);
}

// ------------------------------ launcher -----------------------------------

extern "C" void kernel_launch(void* const* d_in, const int* in_sizes, int n_in,
                              void* d_out, int out_size, void* d_ws, size_t ws_size,
                              hipStream_t stream) {
  (void)n_in; (void)out_size; (void)ws_size;
  const int*   x_rows = (const int*)  d_in[0];
  const int*   x_cols = (const int*)  d_in[1];
  const float* x_vals = (const float*)d_in[2];
  const int*   a_rows = (const int*)  d_in[3];
  const int*   a_cols = (const int*)  d_in[4];
  const float* a_vals = (const float*)d_in[5];
  const float* W      = (const float*)d_in[6];
  const int nnzX = in_sizes[0];
  const int nnzA = in_sizes[3];

  // Workspace carve-up (~155 MB total), 256B-aligned chunks.
  char* ws = (char*)d_ws;
  size_t off = 0;
  auto alloc = [&](size_t b) -> void* {
    void* p = ws + off;
    off = (off + b + 255) & ~(size_t)255;
    return p;
  };
  float*     xw    = (float*)    alloc((size_t)N_NODES * OUT_DIM * sizeof(float));
  unsigned*  x_rp  = (unsigned*) alloc((size_t)(N_NODES + 1) * sizeof(unsigned));
  unsigned*  a_rp  = (unsigned*) alloc((size_t)(N_NODES + 1) * sizeof(unsigned));
  unsigned*  x_cur = (unsigned*) alloc((size_t)N_NODES * sizeof(unsigned));
  unsigned*  a_cur = (unsigned*) alloc((size_t)N_NODES * sizeof(unsigned));
  unsigned*  csums = (unsigned*) alloc((size_t)2 * N_CHUNKS * sizeof(unsigned));
  unsigned*  coffs = (unsigned*) alloc((size_t)2 * N_CHUNKS * sizeof(unsigned));
  long long* x_ent = (long long*)alloc((size_t)nnzX * sizeof(long long));
  long long* a_ent = (long long*)alloc((size_t)nnzA * sizeof(long long));

  // 1) zero per-row counters (deterministic per call; counts live in cursors)
  zero_u32<<<(N_NODES + 1023) / 1024, 1024, 0, stream>>>(x_cur, N_NODES);
  zero_u32<<<(N_NODES + 1023) / 1024, 1024, 0, stream>>>(a_cur, N_NODES);

  // 2) histogram nnz per row
  count_rows<<<(nnzX + 255) / 256, 256, 0, stream>>>(x_rows, x_cur, nnzX);
  count_rows<<<(nnzA + 255) / 256, 256, 0, stream>>>(a_rows, a_cur, nnzA);

  // 3) exclusive scan: WMMA triangular-matmul tiles -> sums -> finalize
  scan_partial_k<<<2 * N_CHUNKS, 32, 0, stream>>>(x_cur, x_rp, a_cur, a_rp,
                                                  csums, N_NODES, N_CHUNKS);
  scan_sums<<<1, 512, 0, stream>>>(csums, coffs, x_rp, a_rp, N_NODES, N_CHUNKS);
  scan_finalize<<<2 * N_CHUNKS, SCAN_TILE, 0, stream>>>(x_rp, x_cur, a_rp, a_cur,
                                                        coffs, N_NODES, N_CHUNKS);

  // 4) permute COO into CSR entry arrays
  scatter_csr<<<(nnzX + 255) / 256, 256, 0, stream>>>(x_rows, x_cols, x_vals,
                                                      x_cur, x_ent, nnzX);
  scatter_csr<<<(nnzA + 255) / 256, 256, 0, stream>>>(a_rows, a_cols, a_vals,
                                                      a_cur, a_ent, nnzA);

  // 5) SpMM1 with TDM-staged 256KB W slabs in LDS (one WGP per block)
  (void)hipFuncSetAttribute(reinterpret_cast<const void*>(spmm1_xw),
                            hipFuncAttributeMaxDynamicSharedMemorySize, LDS_BYTES);
  spmm1_xw<<<SLABS * 128, 1024, LDS_BYTES, stream>>>(x_rp, x_ent, W, xw, N_NODES);

  // 6) SpMM2 + fused ReLU, gathers served from L2-resident xw
  spmm2_relu<<<(N_NODES + 7) / 8, 256, 0, stream>>>(a_rp, a_ent, xw,
                                                    (float*)d_out, N_NODES);
}